// LocalAttention_84473416778159
// MI455X (gfx1250) — compile-verified
//
#include <hip/hip_runtime.h>
#include <math.h>

// ---------------------------------------------------------------------------
// Local windowed attention with RoPE for MI455X (gfx1250), f32 WMMA path.
//   q,k,v: (4,8,4096,64) f32.  WINDOW=128, look_backward=1 -> J=256 keys/window.
//   sim = Qr(128x64) . Kr^T(64x256)  (WMMA f32 16x16x4)
//   softmax with causal+pad mask, out = P(128x256) . V(256x64) (WMMA)
// ---------------------------------------------------------------------------

typedef __attribute__((ext_vector_type(2))) float v2f;
typedef __attribute__((ext_vector_type(8))) float v8f;

#define WINDOW 128
#define JWIN   256
#define DHEAD  64
#define NSEQ   4096
#define NWIN   (NSEQ / WINDOW)   // 32
#define BHTOT  32                // 4*8

#define SK_STRIDE 68             // 256B row + 16B pad -> bank step 4/lane
#define SP_STRIDE 260            // 1040B row -> conflict-free C stores / A loads

// LDS layout (float offsets)
#define OFF_K 0
#define OFF_V (OFF_K + JWIN * SK_STRIDE)     // 17408
#define OFF_Q (OFF_V + JWIN * SK_STRIDE)     // 34816
#define OFF_P (OFF_Q + WINDOW * SK_STRIDE)   // 43520
#define OFF_R (OFF_P + WINDOW * SP_STRIDE)   // 76800
#define LDS_FLOATS (OFF_R + 8 * 16)          // 76928 -> 307712 bytes (<=320KB)

__global__ __launch_bounds__(256, 1)
void local_attn_rope_kernel(const float* __restrict__ q,
                            const float* __restrict__ k,
                            const float* __restrict__ v,
                            float* __restrict__ out) {
  extern __shared__ float smem[];
  float* sK = smem + OFF_K;
  float* sV = smem + OFF_V;
  float* sQ = smem + OFF_Q;
  float* sP = smem + OFF_P;
  float* sR = smem + OFF_R;

  const int w   = blockIdx.x;           // window index 0..31
  const int bh  = blockIdx.y;           // fused batch*head 0..31
  const int tid = threadIdx.x;

  const size_t bhBase = (size_t)bh * NSEQ * DHEAD;

  // inv_freq[i] = 10000^(-i/32) = exp2(-i * log2(10000)/32)
  const float kLog2InvFreq = 13.287712379549449f / 32.0f;

  // ---- Stage K (RoPE), V into LDS: rows j=0..255, global row (w-1)*128 + j
  for (int idx = tid; idx < JWIN * DHEAD; idx += 256) {
    const int j = idx >> 6;
    const int d = idx & 63;
    const int grow = (w - 1) * WINDOW + j;
    float kv, kpart, vv;
    if (grow < 0) {               // pad region (window 0, j<128): masked later
      kv = 0.0f; kpart = 0.0f; vv = 0.0f;
    } else {
      const float* kb = k + bhBase + (size_t)grow * DHEAD;
      kv    = kb[d];
      kpart = kb[d ^ 32];         // rotate_half partner
      vv    = v[bhBase + (size_t)grow * DHEAD + d];
    }
    const float freq = __builtin_amdgcn_exp2f(-(float)(d & 31) * kLog2InvFreq);
    float s, c;
    __sincosf((float)j * freq, &s, &c);
    const float rot = (d < 32) ? -kpart : kpart;
    sK[j * SK_STRIDE + d] = kv * c + rot * s;
    sV[j * SK_STRIDE + d] = vv;
  }
  // ---- Stage Q (RoPE at position 128+i, pre-scaled by D^-0.5 = 1/8)
  for (int idx = tid; idx < WINDOW * DHEAD; idx += 256) {
    const int i = idx >> 6;
    const int d = idx & 63;
    const float* qb = q + bhBase + (size_t)(w * WINDOW + i) * DHEAD;
    const float val  = qb[d];
    const float part = qb[d ^ 32];
    const float freq = __builtin_amdgcn_exp2f(-(float)(d & 31) * kLog2InvFreq);
    float s, c;
    __sincosf((float)(i + WINDOW) * freq, &s, &c);
    const float rot = (d < 32) ? -part : part;
    sQ[i * SK_STRIDE + d] = 0.125f * (val * c + rot * s);
  }
  __syncthreads();

  const int lane = tid & 31;
  const int wave = tid >> 5;
  const int lo   = lane & 15;     // column / row-within-halfwave selector
  const int hi   = lane >> 4;     // K-pair selector (ISA 16x4 f32 A/B layout)
  const int m0   = wave * 16;     // this wave owns query rows m0..m0+15

  // ---- Phase 1: sim strip (16 x 256) = Q . K^T via v_wmma_f32_16x16x4_f32
  const float MASKV = -3.402823466e38f;
  const float* aQ = sQ + (m0 + lo) * SK_STRIDE + 2 * hi;
  for (int n0 = 0; n0 < JWIN; n0 += 16) {
    v8f acc = {};
    const float* bK = sK + (n0 + lo) * SK_STRIDE + 2 * hi;
#pragma unroll
    for (int d0 = 0; d0 < DHEAD; d0 += 4) {
      v2f a = *(const v2f*)(aQ + d0);
      v2f b = *(const v2f*)(bK + d0);
      acc = __builtin_amdgcn_wmma_f32_16x16x4_f32(false, a, false, b,
                                                  (short)0, acc, false, false);
    }
    const int jcol = n0 + lo;     // key column of this lane's C elements
#pragma unroll
    for (int vv = 0; vv < 8; ++vv) {
      const int irow = m0 + vv + 8 * hi;   // local query index 0..127
      const bool masked = (jcol > irow + WINDOW) || (w == 0 && jcol < WINDOW);
      sP[irow * SP_STRIDE + jcol] = masked ? MASKV : acc[vv];
    }
  }
  asm volatile("s_wait_dscnt 0" ::: "memory");  // cross-lane LDS handoff (same wave)

  // ---- Phase 2: row softmax (2 lanes per row, 128 cols each)
  {
    const int row  = m0 + (lane >> 1);
    const int half = lane & 1;
    float* pr = sP + row * SP_STRIDE + half * 128;
    float mx = -INFINITY;
#pragma unroll 8
    for (int c4 = 0; c4 < 128; c4 += 4) {
      float4 x = *(const float4*)(pr + c4);
      mx = fmaxf(mx, fmaxf(fmaxf(x.x, x.y), fmaxf(x.z, x.w)));
    }
    mx = fmaxf(mx, __shfl_xor(mx, 1, 32));
    float sum = 0.0f;
#pragma unroll 8
    for (int c4 = 0; c4 < 128; c4 += 4) {
      float4 x = *(const float4*)(pr + c4);
      x.x = __expf(x.x - mx); x.y = __expf(x.y - mx);
      x.z = __expf(x.z - mx); x.w = __expf(x.w - mx);
      *(float4*)(pr + c4) = x;
      sum += x.x + x.y + x.z + x.w;
    }
    sum += __shfl_xor(sum, 1, 32);
    if (half == 0) sR[wave * 16 + (lane >> 1)] = 1.0f / sum;
  }
  asm volatile("s_wait_dscnt 0" ::: "memory");

  // ---- Phase 3: out strip (16 x 64) = P . V, accumulate over j
  v8f o0 = {}, o1 = {}, o2 = {}, o3 = {};
  const float* aP = sP + (m0 + lo) * SP_STRIDE + 2 * hi;
#pragma unroll 4
  for (int j0 = 0; j0 < JWIN; j0 += 4) {
    v2f a = *(const v2f*)(aP + j0);
    const float* bV = sV + (j0 + 2 * hi) * SK_STRIDE + lo;
    v2f b0 = { bV[0],  bV[SK_STRIDE +  0] };
    v2f b1 = { bV[16], bV[SK_STRIDE + 16] };
    v2f b2 = { bV[32], bV[SK_STRIDE + 32] };
    v2f b3 = { bV[48], bV[SK_STRIDE + 48] };
    o0 = __builtin_amdgcn_wmma_f32_16x16x4_f32(false, a, false, b0, (short)0, o0, false, false);
    o1 = __builtin_amdgcn_wmma_f32_16x16x4_f32(false, a, false, b1, (short)0, o1, false, false);
    o2 = __builtin_amdgcn_wmma_f32_16x16x4_f32(false, a, false, b2, (short)0, o2, false, false);
    o3 = __builtin_amdgcn_wmma_f32_16x16x4_f32(false, a, false, b3, (short)0, o3, false, false);
  }

  // ---- Scale rows by 1/sum and store
  float rinv[8];
#pragma unroll
  for (int vv = 0; vv < 8; ++vv) rinv[vv] = sR[wave * 16 + vv + 8 * hi];

  float* outBase = out + bhBase + (size_t)(w * WINDOW) * DHEAD;
#pragma unroll
  for (int vv = 0; vv < 8; ++vv) {
    const int irow = m0 + vv + 8 * hi;
    float* orow = outBase + (size_t)irow * DHEAD + lo;
    const float f = rinv[vv];
    orow[0]  = o0[vv] * f;
    orow[16] = o1[vv] * f;
    orow[32] = o2[vv] * f;
    orow[48] = o3[vv] * f;
  }
}

extern "C" void kernel_launch(void* const* d_in, const int* in_sizes, int n_in,
                              void* d_out, int out_size, void* d_ws, size_t ws_size,
                              hipStream_t stream) {
  const float* q = (const float*)d_in[0];
  const float* k = (const float*)d_in[1];
  const float* v = (const float*)d_in[2];
  float* out = (float*)d_out;

  const size_t shmem = (size_t)LDS_FLOATS * sizeof(float);  // 307712 B
  (void)hipFuncSetAttribute(reinterpret_cast<const void*>(local_attn_rope_kernel),
                            hipFuncAttributeMaxDynamicSharedMemorySize, (int)shmem);

  dim3 grid(NWIN, BHTOT);   // (32 windows, 32 batch*head)
  local_attn_rope_kernel<<<grid, 256, shmem, stream>>>(q, k, v, out);
}